// Attn_33054068310077
// MI455X (gfx1250) — compile-verified
//
#include <hip/hip_runtime.h>
#include <math.h>

typedef __attribute__((ext_vector_type(2))) float v2f;
typedef __attribute__((ext_vector_type(8))) float v8f;

#define BB 16
#define SS 2048
#define HH 1024

// ---------------------------------------------------------------------------
// Kernel A: w2[n] = sum_h v[h] * W[h][HH + n]   (v^T x W[:,H:2H], 1x1024)
// V_WMMA_F32_16X16X4_F32. 64 output tiles of 16 cols; 4 waves per tile split
// K=1024 into 256-chunks. v is staged in LDS and the A fragment is built
// branch-free (mask multiply) so the B global loads pipeline cleanly.
// Two accumulators break the WMMA RAW chain.
// Grid: 32 blocks x 256 threads (8 waves) -> 256 waves = 64 tiles * 4.
// ---------------------------------------------------------------------------
__global__ void __launch_bounds__(256)
w2_wmma_kernel(const float* __restrict__ attn_w, const float* __restrict__ v,
               float* __restrict__ w2) {
    __shared__ float vlds[HH];
    __shared__ float red[8][16];
    const int t = threadIdx.x;
    ((float4*)vlds)[t] = ((const float4*)v)[t];    // 256 * 4 = 1024 floats
    __syncthreads();

    const int lane = t & 31;
    const int wv   = t >> 5;                       // wave in block, 0..7
    const int gw   = blockIdx.x * 8 + wv;          // global wave, 0..255
    const int tile = gw >> 2;                      // 0..63 -> n0 = tile*16
    const int ksec = gw & 3;                       // K quarter
    const int mn   = lane & 15;                    // row (A) / col (B) index
    const int kk   = (lane >> 4) * 2;              // K sub-index per ISA layout
    const int n0   = tile * 16;
    const float msk = (mn == 0) ? 1.0f : 0.0f;     // only A row M=0 carries v
    const float* Wc = attn_w + HH + n0 + mn;       // this lane's column base

    v8f acc0 = {}, acc1 = {};
    const int kbase = ksec * 256;
#pragma unroll 4
    for (int k0 = 0; k0 < 256; k0 += 8) {
        {
            const int k = kbase + k0 + kk;
            float2 vv = *(const float2*)&vlds[k];  // ds_load_b64, broadcast
            v2f a, b;
            a.x = msk * vv.x;
            a.y = msk * vv.y;
            b.x = Wc[(size_t)k       * (2 * HH)];
            b.y = Wc[(size_t)(k + 1) * (2 * HH)];
            acc0 = __builtin_amdgcn_wmma_f32_16x16x4_f32(
                false, a, false, b, (short)0, acc0, false, false);
        }
        {
            const int k = kbase + k0 + 4 + kk;
            float2 vv = *(const float2*)&vlds[k];
            v2f a, b;
            a.x = msk * vv.x;
            a.y = msk * vv.y;
            b.x = Wc[(size_t)k       * (2 * HH)];
            b.y = Wc[(size_t)(k + 1) * (2 * HH)];
            acc1 = __builtin_amdgcn_wmma_f32_16x16x4_f32(
                false, a, false, b, (short)0, acc1, false, false);
        }
    }
    // D row M=0 lives in VGPR0 of lanes 0..15 (C/D layout)
    if (lane < 16) red[wv][lane] = acc0[0] + acc1[0];
    __syncthreads();
    if (t < 16) {
        float s = red[0][t] + red[1][t] + red[2][t] + red[3][t];
        w2[(blockIdx.x * 2) * 16 + t] = s;
    } else if (t >= 32 && t < 48) {
        int n = t - 32;
        float s = red[4][n] + red[5][n] + red[6][n] + red[7][n];
        w2[(blockIdx.x * 2 + 1) * 16 + n] = s;
    }
}

// ---------------------------------------------------------------------------
// Kernel B: scores[row] = dot(enc[row, 0:1024], w2)  for row in [0, B*S)
// Bandwidth-bound streaming GEMV: coalesced float4 loads (global_load_b128),
// w2 staged in LDS, wave32 shuffle reduction, next-row prefetch.
// Grid: 512 blocks x 256 threads; 8 waves/block, 8 rows/wave.
// ---------------------------------------------------------------------------
__global__ void __launch_bounds__(256)
score_kernel(const float* __restrict__ enc, const float* __restrict__ w2g,
             float* __restrict__ scores) {
    __shared__ float sw2[HH];
    const int t = threadIdx.x;
    ((float4*)sw2)[t] = ((const float4*)w2g)[t];   // 256 * 4 = 1024 floats
    __syncthreads();

    const int lane = t & 31;
    const int wv   = t >> 5;
    const int rowBase = blockIdx.x * 64 + wv * 8;
    const float4* sw4 = (const float4*)sw2;

    for (int r = 0; r < 8; ++r) {
        const int row = rowBase + r;
        const float4* e4 = (const float4*)(enc + (size_t)row * HH);
        if (r < 7) {
            // prefetch next row: one 128B line per lane covers all 4KB
            __builtin_prefetch((const void*)(e4 + 256 + lane * 8), 0, 0);
        }
        float acc = 0.0f;
#pragma unroll
        for (int i = 0; i < 8; ++i) {
            float4 x = e4[i * 32 + lane];          // contiguous 512B per issue
            float4 w = sw4[i * 32 + lane];
            acc = fmaf(x.x, w.x, acc);
            acc = fmaf(x.y, w.y, acc);
            acc = fmaf(x.z, w.z, acc);
            acc = fmaf(x.w, w.w, acc);
        }
#pragma unroll
        for (int off = 16; off > 0; off >>= 1)
            acc += __shfl_xor(acc, off, 32);       // wave32 reduction
        if (lane == 0) scores[row] = acc;
    }
}

// ---------------------------------------------------------------------------
// Kernel C: out[b, s] = softmax_s(scores[b, s]).  One block per batch row.
// ---------------------------------------------------------------------------
__global__ void __launch_bounds__(256)
softmax_kernel(const float* __restrict__ scores, float* __restrict__ out) {
    __shared__ float rbuf[256];
    const int b = blockIdx.x;
    const int t = threadIdx.x;
    const float* sc = scores + b * SS;

    float xs[8];
    float mx = -INFINITY;
#pragma unroll
    for (int i = 0; i < 8; ++i) {
        xs[i] = sc[i * 256 + t];
        mx = fmaxf(mx, xs[i]);
    }
    rbuf[t] = mx;
    __syncthreads();
    for (int off = 128; off > 0; off >>= 1) {
        if (t < off) rbuf[t] = fmaxf(rbuf[t], rbuf[t + off]);
        __syncthreads();
    }
    mx = rbuf[0];
    __syncthreads();

    float sum = 0.0f;
#pragma unroll
    for (int i = 0; i < 8; ++i) {
        xs[i] = __expf(xs[i] - mx);
        sum += xs[i];
    }
    rbuf[t] = sum;
    __syncthreads();
    for (int off = 128; off > 0; off >>= 1) {
        if (t < off) rbuf[t] += rbuf[t + off];
        __syncthreads();
    }
    const float inv = 1.0f / rbuf[0];
    float* o = out + b * SS;
#pragma unroll
    for (int i = 0; i < 8; ++i) o[i * 256 + t] = xs[i] * inv;
}

// ---------------------------------------------------------------------------
extern "C" void kernel_launch(void* const* d_in, const int* in_sizes, int n_in,
                              void* d_out, int out_size, void* d_ws, size_t ws_size,
                              hipStream_t stream) {
    // inputs: 0=hidden [B,1,H] (unused: softmax shift-invariant)
    //         1=encoder_outputs [B,S,H], 2=attn_w [H,2H],
    //         3=attn_b [H] (unused), 4=v [1,H]
    const float* enc    = (const float*)d_in[1];
    const float* attn_w = (const float*)d_in[2];
    const float* v      = (const float*)d_in[4];

    float* w2     = (float*)d_ws;       // 1024 floats
    float* scores = w2 + HH;            // 32768 floats

    hipLaunchKernelGGL(w2_wmma_kernel, dim3(32), dim3(256), 0, stream,
                       attn_w, v, w2);
    hipLaunchKernelGGL(score_kernel, dim3(512), dim3(256), 0, stream,
                       enc, w2, scores);
    hipLaunchKernelGGL(softmax_kernel, dim3(BB), dim3(256), 0, stream,
                       scores, (float*)d_out);
}